// MeanAbsoluteError_26044681683062
// MI455X (gfx1250) — compile-verified
//
#include <hip/hip_runtime.h>
#include <math.h>

// ---------------------------------------------------------------------------
// Heart-rate MAE pipeline for MI455X (gfx1250, wave32, WMMA).
//
// Pipeline (B=256 rows per input, L=900, N_FFT=90900, bins 2020..9090):
//   A) spec  = rfft900(sig) * H(butterworth)           (plain VALU, LDS row)
//   B) fil   = irfft900(spec)  (zero-padded to stride 912)
//   C) ps[k] = |DFT_90900(fil)[k]|^2  k in [2020,9090] -> f32 WMMA GEMM pair
//   D) per-row argmax + peak interpolation -> bpm[512]
//   E) mean |bpm_pred - bpm_gt| -> scalar
//
// Stage C dominates (~13 GFLOP) on V_WMMA_F32_16X16X4_F32: each wave handles
// 4 row tiles per column tile so one cos/sin generation feeds 8 WMMAs.
// v3: fil rows padded with zeros to 912 floats so the K loop has an even
// trip count (228) — removes the loop peel that forced ~45 accumulator
// v_dual_mov copies per iteration in v2.
// ---------------------------------------------------------------------------

typedef __attribute__((ext_vector_type(2))) float v2f;
typedef __attribute__((ext_vector_type(8))) float v8f;

#define NROWS      512      // 256 preds rows + 256 gts rows
#define B_ROWS     256
#define L_SIG      900
#define FIL_STRIDE 912      // zero-padded row stride (even WMMA K trip count)
#define NBINS      451      // rfft bins of length-900 signal
#define SPEC_STRIDE 452
#define NFFT       90900    // 900 + 2 * (100/2 * 900)
#define K0         2020     // first bin with freq >= 40 bpm (exact)
#define MBINS      7071     // bins 2020..9090 inclusive
#define PS_STRIDE  7072     // padded to multiple of 16
#define ROW_TILES  32       // 512/16
#define COL_TILES  442      // 7072/16
#define RT_PER_WAVE 4       // row tiles per wave (8 waves * 4 = 32 = all rows)
#define TWO_PI_F   6.28318530717958647692f

// ---------------- Stage A: rfft(900) * Butterworth H ------------------------
__global__ void bandpass_spec_kernel(const float* __restrict__ preds,
                                     const float* __restrict__ gts,
                                     float* __restrict__ specRe,
                                     float* __restrict__ specIm) {
  __shared__ float srow[L_SIG];
  const int r = blockIdx.y;
  const float* src = (r < B_ROWS) ? (preds + (size_t)r * L_SIG)
                                  : (gts + (size_t)(r - B_ROWS) * L_SIG);
  for (int i = threadIdx.x; i < L_SIG; i += blockDim.x) srow[i] = src[i];
  __syncthreads();

  const int b = blockIdx.x * blockDim.x + threadIdx.x;
  if (b >= NBINS) return;

  const float C = TWO_PI_F / (float)L_SIG;
  float sre = 0.f, sim = 0.f;
  int m = 0;                       // m = (b*n) mod 900, exact
  for (int n = 0; n < L_SIG; ++n) {
    const float th = C * (float)m;
    const float x = srow[n];
    sre = fmaf(x,  __cosf(th), sre);
    sim = fmaf(x, -__sinf(th), sim);   // rfft: X = sum s * e^{-i th}
    m += b; if (m >= L_SIG) m -= L_SIG;
  }

  // Butterworth magnitude response, order 2 -> power 4; H[0] = 0.
  float H;
  if (b == 0) {
    H = 0.f;
  } else {
    const float f  = (float)b / 30.0f + 1e-12f;   // b * Fs / L, Fs=30
    float r1 = 0.6f / f;  r1 *= r1;  r1 *= r1;    // (lowcut/f)^4
    float r2 = f / 4.0f;  r2 *= r2;  r2 *= r2;    // (f/highcut)^4
    H = rsqrtf(1.f + r1) * rsqrtf(1.f + r2);
  }
  specRe[(size_t)r * SPEC_STRIDE + b] = sre * H;
  specIm[(size_t)r * SPEC_STRIDE + b] = sim * H;
}

// ---------------- Stage B: irfft(900), zero-padded to FIL_STRIDE ------------
__global__ void irfft_kernel(const float* __restrict__ specRe,
                             const float* __restrict__ specIm,
                             float* __restrict__ fil) {
  __shared__ float sre[NBINS];
  __shared__ float sim[NBINS];
  const int r = blockIdx.y;
  for (int i = threadIdx.x; i < NBINS; i += blockDim.x) {
    sre[i] = specRe[(size_t)r * SPEC_STRIDE + i];
    sim[i] = specIm[(size_t)r * SPEC_STRIDE + i];
  }
  __syncthreads();

  const int n = blockIdx.x * blockDim.x + threadIdx.x;
  if (n >= FIL_STRIDE) return;
  if (n >= L_SIG) {                 // zero pad so WMMA K loop is rectangular
    fil[(size_t)r * FIL_STRIDE + n] = 0.f;
    return;
  }

  const float C = TWO_PI_F / (float)L_SIG;
  float acc = 0.f;
  int m = n;                        // m = (b*n) mod 900, starting at b=1
  for (int b = 1; b <= 449; ++b) {
    const float th = C * (float)m;
    acc = fmaf( sre[b], __cosf(th), acc);
    acc = fmaf(-sim[b], __sinf(th), acc);
    m += n; if (m >= L_SIG) m -= L_SIG;
  }
  // Nyquist bin (b=450): cos(pi*n) = +/-1, sin term = 0. DC bin has H=0.
  const float nyq = sre[450] * ((n & 1) ? -1.f : 1.f);
  fil[(size_t)r * FIL_STRIDE + n] = (2.f * acc + nyq) * (1.0f / (float)L_SIG);
}

// ---------------- Stage C: fine-DFT power spectrum via f32 WMMA -------------
// One block = one 16-bin column tile. 8 waves * 4 row tiles each = all 512
// rows. Each wave generates the cos/sin B operand ONCE per K-step (exact
// integer angle reduction m = bin*n mod NFFT) and feeds 8 WMMAs with it.
__global__ void __launch_bounds__(256)
finedft_ps_kernel(const float* __restrict__ fil, float* __restrict__ ps) {
  const int wave  = threadIdx.x >> 5;
  const int lane  = threadIdx.x & 31;
  const int c0    = blockIdx.x << 4;               // column tile base (bin-K0)
  const int trow0 = wave * RT_PER_WAVE;            // first row tile for wave

  const int lm = lane & 15;                        // A: M index / B: N index
  const int lh = lane >> 4;                        // 0 or 1
  const int kk = lh << 1;                          // K offset {0,2}

  // B-operand phase state for this lane's column (frequency bin).
  const int bin  = K0 + c0 + lm;                   // <= 9091 (padded col harmless)
  int m0 = bin * kk;                               // bin*{0,2} < NFFT, exact
  int m1 = m0 + bin;                               // < NFFT
  const int step = bin << 2;                       // 4*bin <= 36364
  const float CF = TWO_PI_F / (float)NFFT;

  // A-operand base for row tile trow0; successive tiles are a constant
  // 16*912 floats apart (58368 bytes -> fits 24-bit instruction offset).
  const float* abase = fil + (size_t)((trow0 << 4) + lm) * FIL_STRIDE + kk;

  v8f accRe[RT_PER_WAVE] = {};
  v8f accIm[RT_PER_WAVE] = {};
#pragma unroll 2
  for (int n = 0; n < FIL_STRIDE; n += 4) {        // 228 steps, even
    const float t0 = CF * (float)m0;
    const float t1 = CF * (float)m1;
    v2f bc, bs;
    bc.x = __cosf(t0); bc.y = __cosf(t1);
    bs.x = __sinf(t0); bs.y = __sinf(t1);
#pragma unroll
    for (int t = 0; t < RT_PER_WAVE; ++t) {
      const v2f a = *(const v2f*)(abase + (size_t)t * (16 * FIL_STRIDE) + n);
      // (neg_a, A, neg_b, B, c_mod, C, reuse_a, reuse_b)
      accRe[t] = __builtin_amdgcn_wmma_f32_16x16x4_f32(
          false, a, false, bc, (short)0, accRe[t], false, false);
      accIm[t] = __builtin_amdgcn_wmma_f32_16x16x4_f32(
          false, a, false, bs, (short)0, accIm[t], false, false);
    }
    m0 += step; if (m0 >= NFFT) m0 -= NFFT;        // single subtract suffices
    m1 += step; if (m1 >= NFFT) m1 -= NFFT;
  }

  // D layout: VGPR v, lanes 0-15 -> M=v, lanes 16-31 -> M=v+8; N = lane&15.
  const int col = c0 + lm;
#pragma unroll
  for (int t = 0; t < RT_PER_WAVE; ++t) {
#pragma unroll
    for (int v = 0; v < 8; ++v) {
      const int row = ((trow0 + t) << 4) + v + (lh << 3);
      const float re = accRe[t][v];
      const float im = accIm[t][v];
      ps[(size_t)row * PS_STRIDE + col] = re * re + im * im;
    }
  }
}

// ---------------- Stage D: per-row argmax + peak interpolation --------------
__global__ void argmax_bpm_kernel(const float* __restrict__ ps,
                                  float* __restrict__ bpm) {
  __shared__ float svals[256];
  __shared__ int   sidx[256];
  const int r = blockIdx.x;
  const float* row = ps + (size_t)r * PS_STRIDE;

  float best = -1.f;
  int   bidx = 0x7fffffff;
  for (int j = threadIdx.x; j < MBINS; j += 256) {
    const float v = row[j];
    if (v > best) { best = v; bidx = j; }          // ascending scan: first max
  }
  svals[threadIdx.x] = best;
  sidx[threadIdx.x]  = bidx;
  __syncthreads();
  for (int s = 128; s > 0; s >>= 1) {
    if (threadIdx.x < s) {
      const float ov = svals[threadIdx.x + s];
      const int   oi = sidx[threadIdx.x + s];
      if (ov > svals[threadIdx.x] ||
          (ov == svals[threadIdx.x] && oi < sidx[threadIdx.x])) {
        svals[threadIdx.x] = ov; sidx[threadIdx.x] = oi;
      }
    }
    __syncthreads();
  }
  if (threadIdx.x == 0) {
    const int idx = sidx[0];
    const float fstep = 1800.0f / (float)NFFT;     // Fs*60/NFFT bpm per bin
    float out;
    if (idx <= 0) {
      out = (float)K0 * fstep;
    } else if (idx >= MBINS - 1) {
      out = (float)(K0 + MBINS - 1) * fstep;
    } else {
      const float x0 = row[idx - 1], x1 = row[idx], x2 = row[idx + 1];
      const float d1 = x1 - x0, d2 = x1 - x2;
      const float mn = fminf(d1, d2), mx = fmaxf(d1, d2);
      float off = (1.f - mn / mx) * fstep;
      if (d2 > d1) off = -off;
      out = (float)(K0 + idx) * fstep + off;
    }
    bpm[r] = out;
  }
}

// ---------------- Stage E: MAE over 256 row pairs ---------------------------
__global__ void mae_kernel(const float* __restrict__ bpm,
                           float* __restrict__ out) {
  __shared__ float s[256];
  const int t = threadIdx.x;
  s[t] = fabsf(bpm[t] - bpm[t + B_ROWS]);
  __syncthreads();
  for (int k = 128; k > 0; k >>= 1) {
    if (t < k) s[t] += s[t + k];
    __syncthreads();
  }
  if (t == 0) out[0] = s[0] * (1.0f / (float)B_ROWS);
}

// ---------------------------------------------------------------------------
extern "C" void kernel_launch(void* const* d_in, const int* in_sizes, int n_in,
                              void* d_out, int out_size, void* d_ws, size_t ws_size,
                              hipStream_t stream) {
  (void)in_sizes; (void)n_in; (void)out_size; (void)ws_size;
  const float* preds = (const float*)d_in[0];
  const float* gts   = (const float*)d_in[1];
  float* out = (float*)d_out;

  // Workspace layout (floats). Total ~18.3 MB.
  float* ws     = (float*)d_ws;
  float* specRe = ws;                                       // 512*452
  float* specIm = specRe + (size_t)NROWS * SPEC_STRIDE;     // 512*452
  float* fil    = specIm + (size_t)NROWS * SPEC_STRIDE;     // 512*912
  float* ps     = fil    + (size_t)NROWS * FIL_STRIDE;      // 512*7072
  float* bpm    = ps     + (size_t)NROWS * PS_STRIDE;       // 512

  // A: bandpass spectrum (451 bins per row)
  bandpass_spec_kernel<<<dim3(2, NROWS), dim3(256), 0, stream>>>(
      preds, gts, specRe, specIm);
  // B: inverse rfft back to 900 samples (+ zero pad to 912)
  irfft_kernel<<<dim3(4, NROWS), dim3(256), 0, stream>>>(specRe, specIm, fil);
  // C: fine-DFT power spectrum via f32 WMMA, one block per column tile
  finedft_ps_kernel<<<dim3(COL_TILES), dim3(256), 0, stream>>>(fil, ps);
  // D: per-row argmax + interpolation
  argmax_bpm_kernel<<<dim3(NROWS), dim3(256), 0, stream>>>(ps, bpm);
  // E: scalar MAE
  mae_kernel<<<dim3(1), dim3(256), 0, stream>>>(bpm, out);
}